// GutzwillerWaveFunction_36077725287043
// MI455X (gfx1250) — compile-verified
//
#include <hip/hip_runtime.h>
#include <hip/hip_bf16.h>
#include <math.h>

typedef float v2f __attribute__((ext_vector_type(2)));
typedef float v8f __attribute__((ext_vector_type(8)));

#define NSITES   64
#define ST       65          // LDS row stride in floats (column walks rotate banks)
#define PW       16          // LU panel width (= WMMA tile edge)
#define WPB      2           // waves (= matrices) per block
#define NTHREADS (WPB * 32)
#define EPS_DIAG 1e-8f
#define PI_F     3.14159265358979323846f

// zero-cost compiler scheduling fence; hardware keeps same-wave LDS ops in order
__device__ __forceinline__ void wave_fence() { __builtin_amdgcn_wave_barrier(); }

__global__ __launch_bounds__(NTHREADS)
void gutzwiller_slogdet_wave_kernel(const int* __restrict__ x,
                                    const float* __restrict__ orb,
                                    float* __restrict__ out,
                                    int nmat)
{
    __shared__ float Ash[WPB][NSITES * ST];
    __shared__ int   idxsh[WPB][NSITES];

    const int lane = threadIdx.x & 31;
    const int wv   = threadIdx.x >> 5;
    const int mat  = blockIdx.x * WPB + wv;
    if (mat >= nmat) return;                      // wave-uniform exit only

    float* A   = Ash[wv];
    int*   idx = idxsh[wv];

    // ---- occupied-orbital list via ballot ranks: ups ascending, then downs+64 ----
    const int* xb = x + mat * NSITES;
    const int s0 = xb[lane];
    const int s1 = xb[lane + 32];
    const unsigned u0 = (unsigned)__ballot(s0 == 1);
    const unsigned u1 = (unsigned)__ballot(s1 == 1);
    const unsigned lm = (1u << lane) - 1u;
    const int nUp = __popc(u0) + __popc(u1);
    if (s0 ==  1) idx[__popc(u0 & lm)]                                 = lane;
    if (s1 ==  1) idx[__popc(u0) + __popc(u1 & lm)]                    = lane + 32;
    if (s0 == -1) idx[nUp + __popc(~u0 & lm)]                          = NSITES + lane;
    if (s1 == -1) idx[nUp + __popc(~u0) + __popc(~u1 & lm)]            = NSITES + lane + 32;
    wave_fence();

    // ---- gather rows into LDS (coalesced 256B per row), add EPS on diagonal ----
    for (int r = 0; r < NSITES; ++r) {
        const int row = idx[r];
        float v0 = orb[row * NSITES + lane];
        float v1 = orb[row * NSITES + lane + 32];
        if (r == lane)      v0 += EPS_DIAG;
        if (r == lane + 32) v1 += EPS_DIAG;
        A[r * ST + lane]      = v0;
        A[r * ST + lane + 32] = v1;
    }
    wave_fence();

    // ---- blocked LU with partial pivoting, one wave, no barriers ----
    int swaps = 0;
    for (int kb = 0; kb < NSITES / PW; ++kb) {
        const int pc = kb * PW;

        // panel factorization (columns pc..pc+15)
        for (int j = pc; j < pc + PW; ++j) {
            // pivot: first max of |A[i][j]|, i >= j (isamax semantics)
            const int r0 = j + lane, r1 = j + lane + 32;
            float v0 = (r0 < NSITES) ? fabsf(A[r0 * ST + j]) : -1.0f;
            float v1 = (r1 < NSITES) ? fabsf(A[r1 * ST + j]) : -1.0f;
            float bv = v0; int bi = r0;
            if (v1 > bv) { bv = v1; bi = r1; }
            for (int off = 16; off > 0; off >>= 1) {
                const float ov = __shfl_xor(bv, off, 32);
                const int   oi = __shfl_xor(bi, off, 32);
                if (ov > bv || (ov == bv && oi < bi)) { bv = ov; bi = oi; }
            }
            const int p = bi;                      // uniform across lanes

            if (p != j) {                          // full-row swap, 2 cols per lane
                const float a0 = A[j * ST + lane],      b0 = A[p * ST + lane];
                const float a1 = A[j * ST + lane + 32], b1 = A[p * ST + lane + 32];
                A[j * ST + lane]      = b0;  A[p * ST + lane]      = a0;
                A[j * ST + lane + 32] = b1;  A[p * ST + lane + 32] = a1;
                swaps++;
            }
            wave_fence();

            const float invd = 1.0f / A[j * ST + j];
            const int ce = pc + PW;                // panel column end (exclusive)
            // scale L column + rank-1 update on remaining panel columns (2 rows/lane)
            for (int h = 0; h < 2; ++h) {
                const int i = j + 1 + lane + 32 * h;
                if (i < NSITES) {
                    const float l = A[i * ST + j] * invd;
                    A[i * ST + j] = l;
                    for (int c = j + 1; c < ce; ++c)
                        A[i * ST + c] -= l * A[j * ST + c];
                }
            }
            wave_fence();
        }

        const int c0 = pc + PW;                    // first trailing column
        const int nt = NSITES - c0;                // trailing dimension
        if (nt == 0) break;                        // last panel done

        // U12 = L11^{-1} * A12 : forward substitution over panel rows (lanes = cols)
        for (int i = 0; i < PW - 1; ++i) {
            for (int jj = i + 1; jj < PW; ++jj) {
                const float l = A[(pc + jj) * ST + pc + i];
                for (int c = c0 + lane; c < NSITES; c += 32)
                    A[(pc + jj) * ST + c] -= l * A[(pc + i) * ST + c];
            }
            wave_fence();
        }

        // A22 -= L21 * U12 : 16x16 tiles via V_WMMA_F32_16X16X4_F32, K=16 in 4 chunks
        const int tpr  = nt / 16;
        const int half = lane >> 4;                // 0: lanes 0-15, 1: lanes 16-31
        const int l16  = lane & 15;
        for (int t = 0; t < tpr * tpr; ++t) {      // wave-uniform: EXEC all-ones
            const int row0 = c0 + (t / tpr) * 16;
            const int col0 = c0 + (t % tpr) * 16;
            v8f acc = {};
            for (int kk = 0; kk < 4; ++kk) {
                const int k0 = pc + kk * 4 + 2 * half;
                v2f af, bf;
                // A-frag 16x4: lane m = l16, VGPR v holds K = 2*half + v
                af[0] = A[(row0 + l16) * ST + k0 + 0];
                af[1] = A[(row0 + l16) * ST + k0 + 1];
                // B-frag 4x16: lane n = l16, VGPR v holds K-row = 2*half + v
                bf[0] = A[(k0 + 0) * ST + col0 + l16];
                bf[1] = A[(k0 + 1) * ST + col0 + l16];
                acc = __builtin_amdgcn_wmma_f32_16x16x4_f32(
                    false, af, false, bf, (short)0, acc, false, false);
            }
            // C/D layout: VGPR v -> M = v + 8*half, N = l16 ; subtract product
            for (int v = 0; v < 8; ++v)
                A[(row0 + v + 8 * half) * ST + col0 + l16] -= acc[v];
        }
        wave_fence();
    }

    // ---- slogdet from U diagonal + swap parity (shuffle reduction) ----
    const float d0 = A[lane * ST + lane];
    const float d1 = A[(lane + 32) * ST + lane + 32];
    float s  = logf(fabsf(d0)) + logf(fabsf(d1));
    int   ng = (d0 < 0.0f) + (d1 < 0.0f);
    for (int off = 16; off > 0; off >>= 1) {
        s  += __shfl_xor(s, off, 32);
        ng += __shfl_xor(ng, off, 32);
    }
    if (lane == 0) {
        const int neg = (ng + swaps) & 1;
        out[2 * mat + 0] = s;                      // Re: log|det|
        out[2 * mat + 1] = neg ? PI_F : 0.0f;      // Im: log(sign) = 0 or i*pi
    }
}

extern "C" void kernel_launch(void* const* d_in, const int* in_sizes, int n_in,
                              void* d_out, int out_size, void* d_ws, size_t ws_size,
                              hipStream_t stream) {
    const int*   x   = (const int*)d_in[0];      // (B, 64) int32 spins
    const float* orb = (const float*)d_in[1];    // (128, 64) float32
    float*       out = (float*)d_out;            // (B,) complex64 -> 2*B floats
    const int B = in_sizes[0] / NSITES;
    const int blocks = (B + WPB - 1) / WPB;
    gutzwiller_slogdet_wave_kernel<<<dim3(blocks), dim3(NTHREADS), 0, stream>>>(x, orb, out, B);
}